// AntModel_26499948216648
// MI455X (gfx1250) — compile-verified
//
#include <hip/hip_runtime.h>
#include <stdint.h>

// AntModel forward for MI455X (gfx1250):
//   dest_k = rowwise-argmax(W_k)           (memory-bound: ~151 MB of weights)
//   counts = relu(scatter_add(counts, dest_k))  x3, fused in LDS per batch row
//
// No matmul exists in this computation, so no WMMA. CDNA5-specific paths used:
//   - global_load_async_to_lds_b128 + s_wait_asynccnt (async copy engine)
//   - ds_add_u32 LDS atomics for the scatter
//   - wave32 shuffle reduction for argmax

typedef int v4i __attribute__((ext_vector_type(4)));
typedef __attribute__((address_space(1))) v4i glb_v4i;  // global (__device__) v4i
typedef __attribute__((address_space(3))) v4i lds_v4i;  // LDS (__shared__) v4i

// ---------------------------------------------------------------------------
// Async copy of 4096 ints (16 KB) from global into LDS, 256 threads,
// 16 B per lane per pass (global_load_async_to_lds_b128), then wait.
// ---------------------------------------------------------------------------
__device__ __forceinline__ void copy_dest_table(int* ldsDst, const int* gSrc, int tid) {
#if defined(__gfx1250__) && __has_builtin(__builtin_amdgcn_global_load_async_to_lds_b128)
#pragma unroll
    for (int p = 0; p < 4; ++p) {
        __builtin_amdgcn_global_load_async_to_lds_b128(
            (glb_v4i*)(gSrc + p * 1024 + tid * 4),
            (lds_v4i*)(ldsDst + p * 1024 + tid * 4),
            /*offset=*/0, /*cpol=*/0);
    }
#if __has_builtin(__builtin_amdgcn_s_wait_asynccnt)
    __builtin_amdgcn_s_wait_asynccnt(0);
#else
    asm volatile("s_wait_asynccnt 0" ::: "memory");
#endif
#else
    // Fallback: plain vectorized copy through VGPRs.
    const int4* s = (const int4*)gSrc;
    int4*       d = (int4*)ldsDst;
#pragma unroll
    for (int p = 0; p < 4; ++p) d[p * 256 + tid] = s[p * 256 + tid];
#endif
}

// ---------------------------------------------------------------------------
// Kernel 1: row-wise argmax of W1 (4096x4096), W2 (4096x4096), W3 (4096x1024).
// One wave32 per row; lanes stream float4 (b128 loads), first-index tie-break.
// ---------------------------------------------------------------------------
__global__ void __launch_bounds__(256) ant_argmax_kernel(
    const float* __restrict__ W1, const float* __restrict__ W2,
    const float* __restrict__ W3, int* __restrict__ dests) {
    const int wavesPerBlock = blockDim.x >> 5;
    const int wid  = blockIdx.x * wavesPerBlock + (threadIdx.x >> 5);
    const int lane = threadIdx.x & 31;
    if (wid >= 3 * 4096) return;

    const float* W;
    int ncol, row, base_out;
    if (wid < 4096)       { W = W1; ncol = 4096; row = wid;        base_out = 0;    }
    else if (wid < 8192)  { W = W2; ncol = 4096; row = wid - 4096; base_out = 4096; }
    else                  { W = W3; ncol = 1024; row = wid - 8192; base_out = 8192; }

    const float4* rowp = (const float4*)(W + (size_t)row * (size_t)ncol);
    float best = -__builtin_inff();
    int   bidx = 0;
    const int nIter = ncol >> 7;  // 128 floats per wave per iteration
    for (int k = 0; k < nIter; ++k) {
        float4 v = rowp[k * 32 + lane];
        int idx = k * 128 + lane * 4;
        // In-lane indices increase monotonically -> strict '>' keeps first max.
        if (v.x > best) { best = v.x; bidx = idx;     }
        if (v.y > best) { best = v.y; bidx = idx + 1; }
        if (v.z > best) { best = v.z; bidx = idx + 2; }
        if (v.w > best) { best = v.w; bidx = idx + 3; }
    }
    // wave32 reduction, first-occurrence tie-break (matches jnp.argmax).
#pragma unroll
    for (int off = 16; off > 0; off >>= 1) {
        float ov = __shfl_xor(best, off, 32);
        int   oi = __shfl_xor(bidx, off, 32);
        if (ov > best || (ov == best && oi < bidx)) { best = ov; bidx = oi; }
    }
    if (lane == 0) dests[base_out + row] = bidx;
}

// ---------------------------------------------------------------------------
// Kernel 2: fused 3-layer scatter. One block per batch row; all three layers
// stay resident in LDS (52 KB/block, WGP has 320 KB). Routing tables arrive
// via the async copy engine; scatter uses ds_add_u32 atomics.
// ---------------------------------------------------------------------------
__global__ void __launch_bounds__(256) ant_scatter_kernel(
    const int* __restrict__ x, const int* __restrict__ dests,
    float* __restrict__ out) {
    __shared__ int destBuf[4096];  // 16 KB: current layer's routing table
    __shared__ int accA[4096];     // 16 KB: layer-1 output counts
    __shared__ int accB[4096];     // 16 KB: layer-2 output counts
    __shared__ int accC[1024];     //  4 KB: layer-3 output counts

    const int b = blockIdx.x;
    const int t = threadIdx.x;

    for (int i = t; i < 4096; i += 256) { accA[i] = 0; accB[i] = 0; }
    for (int i = t; i < 1024; i += 256) { accC[i] = 0; }

    // ---- layer 1: x -> accA -------------------------------------------------
    copy_dest_table(destBuf, dests, t);          // overlaps with zero-init above
    __syncthreads();
    const int* xrow = x + (size_t)b * 4096;
    for (int i = t; i < 4096; i += 256) {
        int c = xrow[i];                          // counts in [0, 8)
        if (c) atomicAdd(&accA[destBuf[i]], c);   // ds_add_u32
    }
    __syncthreads();                              // accA done; destBuf free

    // ---- layer 2: relu(accA) -> accB ---------------------------------------
    copy_dest_table(destBuf, dests + 4096, t);
    __syncthreads();
    for (int i = t; i < 4096; i += 256) {
        int c = accA[i];
        c = c > 0 ? c : 0;                        // layer relu
        if (c) atomicAdd(&accB[destBuf[i]], c);
    }
    __syncthreads();

    // ---- layer 3: relu(accB) -> accC (n_out = 1024) -------------------------
    copy_dest_table(destBuf, dests + 8192, t);
    __syncthreads();
    for (int i = t; i < 4096; i += 256) {
        int c = accB[i];
        c = c > 0 ? c : 0;
        if (c) atomicAdd(&accC[destBuf[i]], c);
    }
    __syncthreads();

    // ---- final relu + cast to f32 -------------------------------------------
    float* orow = out + (size_t)b * 1024;
    for (int i = t; i < 1024; i += 256) {
        int c = accC[i];
        orow[i] = (float)(c > 0 ? c : 0);
    }
}

// ---------------------------------------------------------------------------
extern "C" void kernel_launch(void* const* d_in, const int* in_sizes, int n_in,
                              void* d_out, int out_size, void* d_ws, size_t ws_size,
                              hipStream_t stream) {
    (void)in_sizes; (void)n_in; (void)out_size; (void)ws_size;
    const int*   x  = (const int*)  d_in[0];
    const float* W1 = (const float*)d_in[1];
    const float* W2 = (const float*)d_in[2];
    const float* W3 = (const float*)d_in[3];
    int* dests = (int*)d_ws;  // 3 * 4096 ints = 48 KB of workspace

    // 12288 rows, one wave32 per row, 8 waves per 256-thread block.
    ant_argmax_kernel<<<1536, 256, 0, stream>>>(W1, W2, W3, dests);
    // One block per batch element; stream ordering provides the dependency.
    ant_scatter_kernel<<<256, 256, 0, stream>>>(x, dests, (float*)d_out);
}